// EMGaussianMixtureLayer_27857157881992
// MI455X (gfx1250) — compile-verified
//
#include <hip/hip_runtime.h>

// MI455X / gfx1250, wave32. Full-fp32 EM on the matrix pipe
// (V_WMMA_F32_16X16X4_F32):
//   E-step:  q[n,k] = x^T A_k x - 2(A_k mu_k).x + mu^T A_k mu
//            = XX(N x 1024) @ A2^T  +  X @ (-2H)^T  +  c   -> one WMMA GEMM,
//            K in the column dim => per-lane exp, single row-sum shuffle.
//   M-step:  means = P^T @ data ; cov_k = diff^T @ (p_k (*) diff)
// VALU only for the 32 Gauss-Jordan inversions and small reductions.

typedef __attribute__((ext_vector_type(2))) float v2f;
typedef __attribute__((ext_vector_type(8))) float v8f;

constexpr int N = 32768;
constexpr int K = 32;
constexpr int D = 32;
constexpr int NITER = 4;
constexpr int SPLITM = 32;  // n-splits for means GEMM
constexpr int SPLITC = 8;   // n-splits per (k,tile) for covariance GEMM

// ---------------- state init: copy inputs into workspace state ----------------
__global__ void init_state_kernel(const float* __restrict__ means,
                                  const float* __restrict__ cov,
                                  const float* __restrict__ weights,
                                  float* __restrict__ meansS,
                                  float* __restrict__ covS,
                                  float* __restrict__ weightsS) {
  int i = blockIdx.x * blockDim.x + threadIdx.x;
  if (i < K * D) meansS[i] = means[i];
  else if (i < K * D + K * D * D) covS[i - K * D] = cov[i - K * D];
  else if (i < K * D + K * D * D + K)
    weightsS[i - K * D - K * D * D] = weights[i - K * D - K * D * D];
}

// ------- per-component Gauss-Jordan inverse + det + E-step precompute ---------
// one wave per component; thread t owns columns t and t+32 of [A | I].
// Emits, for the WMMA E-step (pair-interleaved for B64 fragment loads):
//   A2P[(dd>>1)*2K + 2k + (dd&1)] = inv_k[dd/32][dd%32]      (dd = 0..1023)
//   Hm2P[(d>>1)*2K + 2k + (d&1)]  = -2 * (inv_k mu_k)[d]
//   cvec[k] = mu_k^T inv_k mu_k ;  scalek[k] = w_k / sqrt(det)
__global__ void invert_kernel(const float* __restrict__ covS,
                              const float* __restrict__ weightsS,
                              const float* __restrict__ meansS,
                              float* __restrict__ A2P,
                              float* __restrict__ Hm2P,
                              float* __restrict__ cvec,
                              float* __restrict__ scalek) {
  __shared__ float M[D][2 * D];
  __shared__ float sMu[D];
  __shared__ float sHmu[D];
  int k = blockIdx.x;
  int t = threadIdx.x;  // 0..31
  for (int r = 0; r < D; ++r) {
    M[r][t] = covS[k * D * D + r * D + t];
    M[r][D + t] = (r == t) ? 1.0f : 0.0f;
  }
  sMu[t] = meansS[k * D + t];
  __syncthreads();
  float det = 1.0f;
  for (int r = 0; r < D; ++r) {
    float piv = M[r][r];
    det *= piv;
    float ip = 1.0f / piv;
    M[r][t] *= ip;
    M[r][D + t] *= ip;
    __syncthreads();
    for (int i = 0; i < D; ++i) {
      if (i == r) continue;
      float f = M[i][r];  // lockstep wave32: all lanes load before any store
      M[i][t] = fmaf(-f, M[r][t], M[i][t]);
      M[i][D + t] = fmaf(-f, M[r][D + t], M[i][D + t]);
    }
    __syncthreads();
  }
  // h_t = sum_e inv[t][e] * mu[e]  (row t of inv lives in M[t][D..2D))
  float h = 0.0f;
  for (int e = 0; e < D; ++e) h = fmaf(M[t][D + e], sMu[e], h);
  Hm2P[(t >> 1) * (2 * K) + 2 * k + (t & 1)] = -2.0f * h;
  sHmu[t] = h * sMu[t];
  // A2P: element (dd = r*32 + t, k)
  for (int r = 0; r < D; ++r) {
    int dd = r * D + t;
    A2P[(dd >> 1) * (2 * K) + 2 * k + (dd & 1)] = M[r][D + t];
  }
  __syncthreads();
  if (t == 0) {
    float c = 0.0f;
    for (int e = 0; e < D; ++e) c += sHmu[e];
    cvec[k] = c;
    scalek[k] = weightsS[k] * rsqrtf(det);
  }
}

// ---------------- E-step (pure WMMA GEMM): probs[n][k] -------------------------
// One wave per 16 points. Q-tile(16 x 32k) accumulated as:
//   phase 1: 512 WMMAs over dd=0..1023 with A-frag = x[d1]*x[d2] (from regs/LDS)
//   phase 2:  16 WMMAs with A-frag = x, B = -2H
// then q += c_k, p = scale*exp(-q/2) per lane (distinct (row,k) per lane),
// one shfl_xor row-sum per row, coalesced normalized store. No divergence.
__global__ void estep_wmma_kernel(const float* __restrict__ data,
                                  const float* __restrict__ A2P,
                                  const float* __restrict__ Hm2P,
                                  const float* __restrict__ cvec,
                                  const float* __restrict__ scalek,
                                  float* __restrict__ probs) {
  __shared__ float sX[16][D];  // this wave's 16 data rows
  int n0 = blockIdx.x * 16;
  int lane = threadIdx.x;
  int half = lane >> 4;  // 0/1
  int l = lane & 15;

  for (int r = 0; r < 16; ++r) sX[r][lane] = data[(n0 + r) * D + lane];
  __syncthreads();

  // lane's own row (A-fragment row = l for both halves): x[4c+2h], x[4c+2h+1]
  v2f xa[8];
#pragma unroll
  for (int c = 0; c < 8; ++c)
    xa[c] = *(const v2f*)&sX[l][4 * c + 2 * half];

  v8f c0 = {}, c1 = {};  // Q cols 0..15 / 16..31

  // ---- phase 1: sum_dd x[d1]*x[d2] * A2[dd,k],  dd = d1*32 + d2 ----
#pragma unroll 1
  for (int d1 = 0; d1 < D; ++d1) {
    float xd1 = sX[l][d1];
    const float* Bp = A2P + (d1 * 16) * (2 * K);  // rows dd = d1*32 ...
#pragma unroll
    for (int c2 = 0; c2 < 8; ++c2) {
      // dd0 = d1*32 + 4*c2 + 2*half (even): pair (dd0, dd0+1) interleaved
      const float* bb = Bp + (2 * c2 + half) * (2 * K);
      v2f a, b0, b1;
      a.x = xd1 * xa[c2].x;
      a.y = xd1 * xa[c2].y;
      b0 = *(const v2f*)(bb + 2 * l);       // (dd0,k=l),(dd0+1,k=l)
      b1 = *(const v2f*)(bb + 32 + 2 * l);  // k = 16+l
      c0 = __builtin_amdgcn_wmma_f32_16x16x4_f32(false, a, false, b0, (short)0,
                                                 c0, false, false);
      c1 = __builtin_amdgcn_wmma_f32_16x16x4_f32(false, a, false, b1, (short)0,
                                                 c1, false, false);
    }
  }
  // ---- phase 2: + sum_d x[d] * (-2H)[d,k] ----
#pragma unroll
  for (int c2 = 0; c2 < 8; ++c2) {
    const float* hb = Hm2P + (2 * c2 + half) * (2 * K);
    v2f b0 = *(const v2f*)(hb + 2 * l);
    v2f b1 = *(const v2f*)(hb + 32 + 2 * l);
    c0 = __builtin_amdgcn_wmma_f32_16x16x4_f32(false, xa[c2], false, b0,
                                               (short)0, c0, false, false);
    c1 = __builtin_amdgcn_wmma_f32_16x16x4_f32(false, xa[c2], false, b1,
                                               (short)0, c1, false, false);
  }

  float cv0 = cvec[l], cv1 = cvec[16 + l];
  float sc0 = scalek[l], sc1 = scalek[16 + l];
#pragma unroll
  for (int j = 0; j < 8; ++j) {
    int row = n0 + j + 8 * half;
    float p0 = sc0 * __expf(-0.5f * (c0[j] + cv0));  // k = l
    float p1 = sc1 * __expf(-0.5f * (c1[j] + cv1));  // k = 16+l
    float s = p0 + p1;                // row-sum over all 32 k (lanes of half)
    s += __shfl_xor(s, 1, 32);
    s += __shfl_xor(s, 2, 32);
    s += __shfl_xor(s, 4, 32);
    s += __shfl_xor(s, 8, 32);
    float is = 1.0f / s;
    probs[row * K + l] = p0 * is;       // coalesced, normalized, no 2nd pass
    probs[row * K + 16 + l] = p1 * is;
  }
}

// ---------------- Nk + weights (deterministic tree reduction) -----------------
__global__ void nk_weights_kernel(const float* __restrict__ probs,
                                  float* __restrict__ Nk,
                                  float* __restrict__ weightsS) {
  __shared__ float red[256];
  int k = blockIdx.x;
  int tid = threadIdx.x;
  float s = 0.0f;
  for (int n = tid; n < N; n += 256) s += probs[n * K + k];
  red[tid] = s;
  __syncthreads();
  for (int off = 128; off > 0; off >>= 1) {
    if (tid < off) red[tid] += red[tid + off];
    __syncthreads();
  }
  if (tid == 0) {
    Nk[k] = red[0];
    weightsS[k] = red[0] * (1.0f / (float)N);
  }
}

// ------------- means GEMM: C[k,d] = sum_n probs[n][k]*data[n][d] (WMMA) -------
__global__ void means_partial_kernel(const float* __restrict__ probs,
                                     const float* __restrict__ data,
                                     float* __restrict__ partM) {
  int tile = blockIdx.x;      // 0..3 : (kTile*2 + dTile)
  int split = blockIdx.y;     // 0..SPLITM-1
  int mt = (tile >> 1) * 16;  // k-row offset
  int nt = (tile & 1) * 16;   // d-col offset
  int lane = threadIdx.x;
  int half = lane >> 4;
  int l = lane & 15;
  int chunk = N / SPLITM;
  int n0 = split * chunk;
  v8f c = {};
  for (int nn = n0; nn < n0 + chunk; nn += 4) {
    int r0 = nn + 2 * half;  // A/B VGPR0 holds Kdim 0/2, VGPR1 holds 1/3
    v2f a, b;
    a.x = probs[r0 * K + mt + l];
    a.y = probs[(r0 + 1) * K + mt + l];
    b.x = data[r0 * D + nt + l];
    b.y = data[(r0 + 1) * D + nt + l];
    c = __builtin_amdgcn_wmma_f32_16x16x4_f32(false, a, false, b, (short)0, c,
                                              false, false);
  }
  float* out = partM + (tile * SPLITM + split) * 256;
#pragma unroll
  for (int j = 0; j < 8; ++j) out[(j + 8 * half) * 16 + l] = c[j];
}

__global__ void means_combine_kernel(const float* __restrict__ partM,
                                     const float* __restrict__ Nk,
                                     float* __restrict__ meansS) {
  int idx = blockIdx.x * blockDim.x + threadIdx.x;  // 0..1023
  int k = idx / D, d = idx % D;
  int tile = (k / 16) * 2 + (d / 16);
  int local = (k % 16) * 16 + (d % 16);
  float s = 0.0f;
  for (int sp = 0; sp < SPLITM; ++sp)
    s += partM[(tile * SPLITM + sp) * 256 + local];
  meansS[idx] = s / Nk[k];
}

// -- covariance scatter: C_k[d,e] = sum_n p[n,k]*diff[n,d]*diff[n,e] (WMMA) ----
// diff w.r.t. OLD means (launched before means_combine overwrites meansS).
__global__ void cov_partial_kernel(const float* __restrict__ probs,
                                   const float* __restrict__ data,
                                   const float* __restrict__ meansOld,
                                   float* __restrict__ partC) {
  int tile = blockIdx.x;   // 0..3
  int split = blockIdx.y;  // 0..SPLITC-1
  int k = blockIdx.z;      // 0..31
  int mt = (tile >> 1) * 16;
  int nt = (tile & 1) * 16;
  int lane = threadIdx.x;
  int half = lane >> 4;
  int l = lane & 15;
  float muA = meansOld[k * D + mt + l];
  float muB = meansOld[k * D + nt + l];
  int chunk = N / SPLITC;
  int n0 = split * chunk;
  v8f c = {};
  for (int nn = n0; nn < n0 + chunk; nn += 4) {
    int r0 = nn + 2 * half;
    float p0 = probs[r0 * K + k];  // uniform across lane group (broadcast)
    float p1 = probs[(r0 + 1) * K + k];
    v2f a, b;
    a.x = data[r0 * D + mt + l] - muA;         // A[M=d,Kdim=n] = diff
    a.y = data[(r0 + 1) * D + mt + l] - muA;
    b.x = (data[r0 * D + nt + l] - muB) * p0;  // B[Kdim=n,N=e] = p*diff
    b.y = (data[(r0 + 1) * D + nt + l] - muB) * p1;
    c = __builtin_amdgcn_wmma_f32_16x16x4_f32(false, a, false, b, (short)0, c,
                                              false, false);
  }
  float* out = partC + ((k * 4 + tile) * SPLITC + split) * 256;
#pragma unroll
  for (int j = 0; j < 8; ++j) out[(j + 8 * half) * 16 + l] = c[j];
}

__global__ void cov_combine_kernel(const float* __restrict__ partC,
                                   const float* __restrict__ Nk,
                                   float* __restrict__ covS) {
  int k = blockIdx.x;
  float ink = 1.0f / Nk[k];
  for (int idx = threadIdx.x; idx < D * D; idx += blockDim.x) {
    int d = idx / D, e = idx % D;
    int tile = (d / 16) * 2 + (e / 16);
    int local = (d % 16) * 16 + (e % 16);
    float s = 0.0f;
    for (int sp = 0; sp < SPLITC; ++sp)
      s += partC[((k * 4 + tile) * SPLITC + sp) * 256 + local];
    covS[k * D * D + idx] = s * ink;
  }
}

// ---------------- final: (means, covariances, weights) concat -----------------
__global__ void finalize_kernel(const float* __restrict__ meansS,
                                const float* __restrict__ covS,
                                const float* __restrict__ weightsS,
                                float* __restrict__ out) {
  int i = blockIdx.x * blockDim.x + threadIdx.x;
  if (i < K * D) out[i] = meansS[i];
  else if (i < K * D + K * D * D) out[i] = covS[i - K * D];
  else if (i < K * D + K * D * D + K)
    out[i] = weightsS[i - K * D - K * D * D];
}

extern "C" void kernel_launch(void* const* d_in, const int* in_sizes, int n_in,
                              void* d_out, int out_size, void* d_ws, size_t ws_size,
                              hipStream_t stream) {
  (void)in_sizes; (void)n_in; (void)out_size; (void)ws_size;
  const float* data = (const float*)d_in[0];
  const float* means0 = (const float*)d_in[1];
  const float* cov0 = (const float*)d_in[2];
  const float* w0 = (const float*)d_in[3];
  float* out = (float*)d_out;

  float* ws = (float*)d_ws;  // ~5.5 MB total, 128B-aligned sub-buffers
  float* meansS = ws;   ws += K * D;
  float* covS = ws;     ws += K * D * D;
  float* weightsS = ws; ws += K;
  float* A2P = ws;      ws += D * D * K;  // pair-interleaved vec(inv_k)
  float* Hm2P = ws;     ws += D * K;      // pair-interleaved -2*inv_k*mu_k
  float* cvec = ws;     ws += K;
  float* scalek = ws;   ws += K;
  float* Nk = ws;       ws += K;
  float* probs = ws;    ws += (size_t)N * K;
  float* partM = ws;    ws += 4 * SPLITM * 256;
  float* partC = ws;    // K*4*SPLITC*256

  int totalState = K * D + K * D * D + K;
  init_state_kernel<<<(totalState + 255) / 256, 256, 0, stream>>>(
      means0, cov0, w0, meansS, covS, weightsS);

  for (int it = 0; it < NITER; ++it) {
    invert_kernel<<<K, D, 0, stream>>>(covS, weightsS, meansS, A2P, Hm2P, cvec,
                                       scalek);
    estep_wmma_kernel<<<N / 16, 32, 0, stream>>>(data, A2P, Hm2P, cvec, scalek,
                                                 probs);
    nk_weights_kernel<<<K, 256, 0, stream>>>(probs, Nk, weightsS);
    means_partial_kernel<<<dim3(4, SPLITM), 32, 0, stream>>>(probs, data, partM);
    // cov reads OLD means -> must precede means_combine (in-stream order)
    cov_partial_kernel<<<dim3(4, SPLITC, K), 32, 0, stream>>>(probs, data,
                                                              meansS, partC);
    means_combine_kernel<<<4, 256, 0, stream>>>(partM, Nk, meansS);
    cov_combine_kernel<<<K, 256, 0, stream>>>(partC, Nk, covS);
  }

  finalize_kernel<<<(totalState + 255) / 256, 256, 0, stream>>>(
      meansS, covS, weightsS, out);
}